// GNNLSTMModel_35442070127333
// MI455X (gfx1250) — compile-verified
//
#include <hip/hip_runtime.h>
#include <hip/hip_bf16.h>
#include <math.h>

// ---------------- problem constants ----------------
#define Bz 256
#define Tz 12
#define NF 32
#define FD 8
#define Hz 256          // hidden = NF*FD
#define H4 1024         // 4*H
#define Cz 32
#define BT (Bz*Tz)      // 3072
#define BH (Bz*Hz)      // 65536
#define KC 8            // k-rows staged per TDM chunk (2 slabs x 32KB LDS)

typedef __attribute__((ext_vector_type(2))) float v2f;
typedef __attribute__((ext_vector_type(8))) float v8f;
typedef __attribute__((ext_vector_type(4))) unsigned int u32x4;
typedef __attribute__((ext_vector_type(8))) int i32x8;
typedef __attribute__((ext_vector_type(4))) int i32x4;

__device__ __forceinline__ float sigf(float x) { return 1.0f / (1.0f + __expf(-x)); }

// ---------------- workspace layout (floats) ----------------
#define WS_XT   0
#define WS_AGG  (WS_XT  + (size_t)Tz*BH)
#define WS_SFW  (WS_AGG + (size_t)Tz*BH)
#define WS_SBW  (WS_SFW + (size_t)Tz*BH)
#define WS_HB   (WS_SBW + (size_t)Tz*BH)
#define WS_LM   (WS_HB  + (size_t)12*2*BH)

// ==================================================================
// Kernel 1: attention + graph softmax + aggregation  (per-row wave)
// ==================================================================
__global__ __launch_bounds__(256)
void k1_prep(const float* __restrict__ features,
             const float* __restrict__ aw,        // [32][8]
             const float* __restrict__ gw,        // [16]
             const float* __restrict__ gb,        // [1]
             const float* __restrict__ w_out,     // [8][8]
             const float* __restrict__ b_out,     // [32]
             const float* __restrict__ w_in,      // [8][8]
             const float* __restrict__ b_in,      // [32]
             float* __restrict__ ws)
{
    __shared__ float smax[NF][FD];
    __shared__ float wmbuf[8][NF][NF];
    __shared__ float cstat[8][2][NF];
    __shared__ float xob[8][NF][FD];

    const int tid  = threadIdx.x;
    const int w    = tid >> 5;
    const int lane = tid & 31;

    if (tid < NF) {
        float v[FD], mx = -1e30f;
        for (int d = 0; d < FD; ++d) { v[d] = aw[tid*FD + d]; mx = fmaxf(mx, v[d]); }
        float s = 0.f;
        for (int d = 0; d < FD; ++d) { v[d] = __expf(v[d] - mx); s += v[d]; }
        for (int d = 0; d < FD; ++d) smax[tid][d] = 1.0f + v[d] / s;
    }
    __syncthreads();

    const int row = blockIdx.x * 8 + w;
    const int i   = lane;

    float f[FD], amax = 0.f;
    const float* frow = features + (size_t)row*NF*FD + i*FD;
    for (int d = 0; d < FD; ++d) {
        float raw = frow[d];
        amax = fmaxf(amax, fabsf(raw));
        f[d] = raw * smax[i][d];
    }
    float pa = 0.f, pb = 0.f;
    for (int d = 0; d < FD; ++d) { pa += f[d]*gw[d]; pb += f[d]*gw[FD+d]; }
    const float gbias = gb[0];

    float mreg[NF];
    for (int j = 0; j < NF; ++j) {
        float paj = __shfl(pa, j, 32);
        float pbj = __shfl(pb, j, 32);
        float v = (i < j) ? (paj + pb + gbias) : (pa + pbj + gbias);
        if (i == j) v = 0.f;
        mreg[j] = v;
        wmbuf[w][i][j] = v;
    }
    __syncthreads();

    {
        float cmax = -1e30f;
        for (int r = 0; r < NF; ++r) cmax = fmaxf(cmax, wmbuf[w][r][lane]);
        float cs = 0.f;
        for (int r = 0; r < NF; ++r) cs += __expf(wmbuf[w][r][lane] - cmax);
        cstat[w][0][lane] = cmax;
        cstat[w][1][lane] = cs;
    }
    __syncthreads();

    float g[NF];
    for (int j = 0; j < NF; ++j)
        g[j] = __expf(mreg[j] - cstat[w][0][j]) / cstat[w][1][j];

    float xo[FD];
    for (int d = 0; d < FD; ++d) {
        float s = b_out[i];
        for (int e = 0; e < FD; ++e) s += f[e]*w_out[e*FD + d];
        xo[d] = s;
        xob[w][i][d] = s;
    }
    __syncthreads();

    float gx[FD];
    for (int d = 0; d < FD; ++d) gx[d] = 0.f;
    for (int j = 0; j < NF; ++j) {
        float gj = g[j];
        for (int d = 0; d < FD; ++d) gx[d] += gj * xob[w][j][d];
    }
    float agg[FD];
    for (int d = 0; d < FD; ++d) {
        float s = b_in[i];
        for (int e = 0; e < FD; ++e) s += gx[e]*w_in[e*FD + d];
        agg[d] = s;
    }

    const int b = row / Tz, t = row % Tz;
    float* XT  = ws + WS_XT;
    float* AGG = ws + WS_AGG;
    const size_t base = (size_t)t*BH + (size_t)b*Hz + i*FD;
    for (int d = 0; d < FD; ++d) { XT[base + d] = f[d]; AGG[base + d] = agg[d]; }

    for (int off = 16; off > 0; off >>= 1) amax = fmaxf(amax, __shfl_xor(amax, off, 32));
    if (lane == 0) (ws + WS_LM)[row] = (amax > 0.f) ? 1.0f : 0.0f;
}

// ==================================================================
// TDM: DMA one [KC x 1024] f32 slab (row stride 1024) into LDS.
// Descriptor per CDNA5 ISA ch.8 (D# groups). 2D tile, no padding,
// no multicast, data_size=4B, type=2 ("image").
// This toolchain's builtin takes 6 args:
//   (u32x4 g0, i32x8 g1, i32x4 g2, i32x4 g3, i32x8 extra, i32 cpol)
// groups 2/3 (and the extra descriptor words) are unused for 2D tiles.
// ==================================================================
__device__ __forceinline__ void tdm_load_slab(const float* src, unsigned lds_off_bytes)
{
    unsigned long long ga = (unsigned long long)(uintptr_t)src;
    u32x4 g0;
    g0[0] = 1u;                                           // count=1 (valid, user mode)
    g0[1] = lds_off_bytes;                                // lds_addr (bytes)
    g0[2] = (unsigned)(ga & 0xFFFFFFFFu);                 // global_addr[31:0]
    g0[3] = (unsigned)((ga >> 32) & 0x01FFFFFFu)          // global_addr[56:32]
          | (2u << 30);                                   // type = 2
    i32x8 g1;
    g1[0] = (int)(2u << 16);        // workgroup_mask=0, data_size=2 (4 bytes)
    g1[1] = (int)(1024u << 16);     // tensor_dim0[15:0] = 1024 (bits 63:48)
    g1[2] = (int)((unsigned)KC << 16); // tensor_dim0 hi = 0; tensor_dim1 = KC
    g1[3] = (int)(1024u << 16);     // tensor_dim1 hi = 0; tile_dim0 = 1024
    g1[4] = KC;                     // tile_dim1 = KC, tile_dim2 = 0
    g1[5] = 1024;                   // tensor_dim0_stride[31:0] = 1024 elements
    g1[6] = 0;                      // stride0 hi, tensor_dim1_stride lo
    g1[7] = 0;
    i32x4 zz4 = { 0, 0, 0, 0 };     // groups 2/3 unused (2D tile)
    i32x8 zz8 = { 0, 0, 0, 0, 0, 0, 0, 0 };
    __builtin_amdgcn_tensor_load_to_lds(g0, g1, zz4, zz4, zz8, 0);
}

// ==================================================================
// Kernel 2: LSTM chains via V_WMMA_F32_16X16X4_F32 with TDM-staged
// K/R slabs in LDS.
// grid = 13 blocks x 1024 threads (32 waves).
//   block 0   : forward chain (12 steps), h_t -> SFW[t]
//   block 1+j : backward chain j (j+1 steps), final h -> SBW[j]
// Each wave owns 8 tile-groups (one m-tile, 8 n'-tiles) and processes
// them in pairs so 2x4x8 accumulator VGPRs stay live while the
// k-chunk loop (block-uniform, TDM-staged) runs outermost.
// ==================================================================
__device__ __forceinline__ void lstm_step(
    const float* __restrict__ x,      // [B][H]
    const float* __restrict__ hprev,  // [B][H]
    const float* __restrict__ Kmat,   // [H][4H]
    const float* __restrict__ Rmat,   // [H][4H]
    const float* __restrict__ bias,   // [4H]
    const float* __restrict__ cprev,  // [B][H] or nullptr (first step)
    float* __restrict__ hout,         // [B][H]
    int lane, int wave, int tid,
    float (*slabK)[H4], float (*slabR)[H4],
    unsigned ldsK, unsigned ldsR)
{
    const int l16 = lane & 15;
    const int khi = lane >> 4;
    for (int gp = 0; gp < 4; ++gp) {
        const int grp   = wave * 8 + gp * 2;       // even -> pair shares m-tile
        const int mt    = grp >> 4;
        const int ncol0 = (grp & 15) * 16 + l16;   // group A n' column
        const int m     = mt * 16 + l16;           // batch row for A frags
        v8f acc[2][4] = {};
        for (int kc = 0; kc < Hz; kc += KC) {
            __syncthreads();                       // prior slab fully consumed
            if (tid < 32) {
                tdm_load_slab(Kmat + (size_t)kc * H4, ldsK);
                tdm_load_slab(Rmat + (size_t)kc * H4, ldsR);
                __builtin_amdgcn_s_wait_tensorcnt(0);
            }
            __syncthreads();                       // slab visible to all waves
            #pragma unroll
            for (int k0 = 0; k0 < KC; k0 += 4) {
                const int kl = k0 + 2 * khi;       // slab-local row (even)
                const int kg = kc + kl;            // global k
                v2f ax = *(const v2f*)(x     + (size_t)m * Hz + kg);
                v2f ah = *(const v2f*)(hprev + (size_t)m * Hz + kg);
                #pragma unroll
                for (int g2 = 0; g2 < 2; ++g2) {
                    const int nc = ncol0 + 16 * g2;
                    #pragma unroll
                    for (int gate = 0; gate < 4; ++gate) {
                        const int col = gate * 256 + nc;
                        v2f bk; bk[0] = slabK[kl][col]; bk[1] = slabK[kl + 1][col];
                        acc[g2][gate] = __builtin_amdgcn_wmma_f32_16x16x4_f32(
                            false, ax, false, bk, (short)0, acc[g2][gate], false, false);
                        v2f br; br[0] = slabR[kl][col]; br[1] = slabR[kl + 1][col];
                        acc[g2][gate] = __builtin_amdgcn_wmma_f32_16x16x4_f32(
                            false, ah, false, br, (short)0, acc[g2][gate], false, false);
                    }
                }
            }
        }
        // cell nonlinearity.  C/D layout: VGPR r, lane: M = mt*16 + r + 8*khi, N = nc
        #pragma unroll
        for (int g2 = 0; g2 < 2; ++g2) {
            const int nc = ncol0 + 16 * g2;
            const float bi = bias[        nc];
            const float bf = bias[256   + nc];
            const float bg = bias[512   + nc];
            const float bo = bias[768   + nc];
            #pragma unroll
            for (int r = 0; r < 8; ++r) {
                const int mm = mt * 16 + r + 8 * khi;
                const float zi = acc[g2][0][r] + bi;
                const float zf = acc[g2][1][r] + bf;
                const float zg = acc[g2][2][r] + bg;
                const float zo = acc[g2][3][r] + bo;
                const float cp = cprev ? cprev[(size_t)mm * Hz + nc] : 0.0f;
                const float c  = sigf(zf) * cp + sigf(zi) * tanhf(zg);
                hout[(size_t)mm * Hz + nc] = sigf(zo) * tanhf(c);
            }
        }
    }
}

__global__ __launch_bounds__(1024)
void k2_lstm(const float* __restrict__ Kmat,
             const float* __restrict__ Rmat,
             const float* __restrict__ bias,
             float* __restrict__ ws)
{
    __shared__ float slabK[KC][H4];     // 32 KB
    __shared__ float slabR[KC][H4];     // 32 KB

    const int tid  = threadIdx.x;
    const int lane = tid & 31;
    const int wave = tid >> 5;
    const unsigned ldsK = (unsigned)(uintptr_t)&slabK[0][0];
    const unsigned ldsR = (unsigned)(uintptr_t)&slabR[0][0];

    float* XT  = ws + WS_XT;
    float* AGG = ws + WS_AGG;
    float* SFW = ws + WS_SFW;
    float* SBW = ws + WS_SBW;
    float* HB  = ws + WS_HB;

    if (blockIdx.x == 0) {
        for (int t = 0; t < Tz; ++t) {
            const float* xp = XT + (size_t)t*BH;
            const float* hp = (t == 0) ? AGG : (SFW + (size_t)(t-1)*BH);
            const float* cp = (t == 0) ? nullptr : hp;
            float* out = SFW + (size_t)t*BH;
            lstm_step(xp, hp, Kmat, Rmat, bias, cp, out,
                      lane, wave, tid, slabK, slabR, ldsK, ldsR);
            __syncthreads();
        }
    } else {
        const int j = blockIdx.x - 1;            // 0..11
        float* buf0 = HB + (size_t)(2*j    )*BH;
        float* buf1 = HB + (size_t)(2*j + 1)*BH;
        const float* prev = AGG + (size_t)j*BH;
        for (int s = 0; s <= j; ++s) {
            const float* xp = XT + (size_t)(j - s)*BH;
            const float* cp = (s == 0) ? nullptr : prev;
            float* out = (s == j) ? (SBW + (size_t)j*BH) : ((s & 1) ? buf1 : buf0);
            lstm_step(xp, prev, Kmat, Rmat, bias, cp, out,
                      lane, wave, tid, slabK, slabR, ldsK, ldsR);
            __syncthreads();
            prev = out;
        }
    }
}

// ==================================================================
// Kernel 3: output projection + double softmax + losses
// ==================================================================
__global__ __launch_bounds__(256)
void k3_out(const float* __restrict__ ow,    // [512][32]
            const float* __restrict__ ob,    // [32]
            const float* __restrict__ label, // [B*T]
            const float* __restrict__ mask1, // [B*T][32]
            const float* __restrict__ mask2, // [B*T][32]
            const float* __restrict__ ws,
            float* __restrict__ d_out)
{
    const int tid  = threadIdx.x;
    const int w    = tid >> 5;
    const int c    = tid & 31;
    const int row  = blockIdx.x * 8 + w;
    const int b = row / Tz, t = row % Tz;

    const float* SFW = ws + WS_SFW;
    const float* SBW = ws + WS_SBW;
    const float* srow_f = SFW + (size_t)t*BH + (size_t)b*Hz;
    const float* srow_b = SBW + (size_t)t*BH + (size_t)b*Hz;

    float acc = ob[c];
    for (int k = 0; k < Hz; ++k) acc += srow_f[k] * ow[k*Cz + c];
    for (int k = 0; k < Hz; ++k) acc += srow_b[k] * ow[(Hz + k)*Cz + c];

    float mx = acc;
    for (int o = 16; o > 0; o >>= 1) mx = fmaxf(mx, __shfl_xor(mx, o, 32));
    float e = __expf(acc - mx);
    float sm = e;
    for (int o = 16; o > 0; o >>= 1) sm += __shfl_xor(sm, o, 32);
    float p = e / sm;
    float mx2 = p;
    for (int o = 16; o > 0; o >>= 1) mx2 = fmaxf(mx2, __shfl_xor(mx2, o, 32));
    float e2 = __expf(p - mx2);
    float sm2 = e2;
    for (int o = 16; o > 0; o >>= 1) sm2 += __shfl_xor(sm2, o, 32);
    float p2 = e2 / sm2;

    const float lm = (ws + WS_LM)[row];
    const float pred = lm * p2;
    d_out[(size_t)row*Cz + c] = pred;

    float s1 = mask1[(size_t)row*Cz + c] * pred;
    float s2 = mask2[(size_t)row*Cz + c] * pred;
    for (int o = 16; o > 0; o >>= 1) { s1 += __shfl_xor(s1, o, 32); s2 += __shfl_xor(s2, o, 32); }

    if (c == 0) {
        const float lab = label[row];
        const float inv = 1.0f / (float)BT;
        atomicAdd(&d_out[(size_t)BT*Cz + 0], -(lab*s1 + (1.0f - lab)*s1) * inv);
        atomicAdd(&d_out[(size_t)BT*Cz + 2], -(lab*s2) * inv);
    }
}

__global__ void k0_init(float* __restrict__ d_out)
{
    if (threadIdx.x == 0) {
        d_out[(size_t)BT*Cz + 0] = 0.0f;
        d_out[(size_t)BT*Cz + 1] = 0.0f;   // reference returns literal 0.0
        d_out[(size_t)BT*Cz + 2] = 0.0f;
    }
}

// ==================================================================
extern "C" void kernel_launch(void* const* d_in, const int* in_sizes, int n_in,
                              void* d_out, int out_size, void* d_ws, size_t ws_size,
                              hipStream_t stream) {
    const float* features = (const float*)d_in[0];
    const float* label    = (const float*)d_in[2];
    const float* mask1    = (const float*)d_in[3];
    const float* mask2    = (const float*)d_in[4];
    const float* aw       = (const float*)d_in[5];
    const float* gw       = (const float*)d_in[6];
    const float* gb       = (const float*)d_in[7];
    const float* w_out    = (const float*)d_in[8];
    const float* b_out    = (const float*)d_in[9];
    const float* w_in     = (const float*)d_in[10];
    const float* b_in     = (const float*)d_in[11];
    const float* lstm_k   = (const float*)d_in[12];
    const float* lstm_r   = (const float*)d_in[13];
    const float* lstm_b   = (const float*)d_in[14];
    const float* ow       = (const float*)d_in[15];
    const float* ob       = (const float*)d_in[16];
    float* out = (float*)d_out;
    float* ws  = (float*)d_ws;

    k0_init<<<1, 64, 0, stream>>>(out);
    k1_prep<<<BT/8, 256, 0, stream>>>(features, aw, gw, gb, w_out, b_out, w_in, b_in, ws);
    k2_lstm<<<13, 1024, 0, stream>>>(lstm_k, lstm_r, lstm_b, ws);
    k3_out<<<BT/8, 256, 0, stream>>>(ow, ob, label, mask1, mask2, ws, out);
}